// SemanticKDLoss_49881750176128
// MI455X (gfx1250) — compile-verified
//
#include <hip/hip_runtime.h>
#include <math.h>

// ---------------------------------------------------------------------------
// Semantic-KD loss for MI455X (gfx1250, wave32).
// 134 MB of fp32 inputs -> ~5.8us HBM roofline; each input byte is read from
// HBM exactly once. Top-500 via 4-pass radix select on LDS-resident teacher
// slice; softmax/KL via one fused pass. All block-level float reductions use
// V_WMMA_F32_16X16X4_F32 with a ones B-matrix (layout-invariant) so reduction
// order is fixed -> deterministic output (no float atomics).
// ---------------------------------------------------------------------------

#define TOT 16368
#define KCAP 500
#define NORM_OVER_B (0.1f / 1024.0f)
#define NPART 10240 /* 1024 rows * 5 large + 1024 rows * 5 small */

typedef __attribute__((ext_vector_type(2))) float v2f;
typedef __attribute__((ext_vector_type(8))) float v8f;

// Monotonic float->uint key: bigger float => bigger key.
__device__ __forceinline__ unsigned fkey(float f) {
  unsigned u = __float_as_uint(f);
  return u ^ (unsigned)(((int)u >> 31) | (int)0x80000000);
}

// ---------------------------------------------------------------------------
// Deterministic 512-thread block sum using V_WMMA_F32_16X16X4_F32.
// 512 partials -> LDS. Wave 0 builds A (16x4 fp32, ISA layout: lanes 0-15 hold
// M=lane with VGPR0=K0,VGPR1=K1; lanes 16-31 hold M=lane-16 with K2,K3),
// B = ones (all-ones is invariant to the B layout), and accumulates 8 chunks
// of 64 into C. D[m,n] = sum_k A[m,k]; each lane then folds its 8 C VGPRs
// (M = vgpr + (lane>=16 ? 8 : 0)) and one shfl_xor(16) joins the two halves.
// ---------------------------------------------------------------------------
__device__ __forceinline__ float blockSumWMMA(float v, float* red, int tid) {
  red[tid] = v;
  __syncthreads();
  if (tid < 32) {
    const int lane = tid;
    v8f acc = {0.f, 0.f, 0.f, 0.f, 0.f, 0.f, 0.f, 0.f};
    v2f b;
    b.x = 1.0f;
    b.y = 1.0f;
#pragma unroll
    for (int c = 0; c < 8; ++c) {
      const float* base = red + c * 64;
      v2f a;
      if (lane < 16) {
        a.x = base[4 * lane + 0];  // M=lane, K=0
        a.y = base[4 * lane + 1];  // M=lane, K=1
      } else {
        a.x = base[4 * (lane - 16) + 2];  // M=lane-16, K=2
        a.y = base[4 * (lane - 16) + 3];  // M=lane-16, K=3
      }
      acc = __builtin_amdgcn_wmma_f32_16x16x4_f32(
          false, a, false, b, (short)0, acc, false, false);
    }
    float s = acc[0] + acc[1] + acc[2] + acc[3] + acc[4] + acc[5] + acc[6] +
              acc[7];                 // sum over this half's 8 M rows
    s += __shfl_xor(s, 16, 32);       // join M=0..7 with M=8..15 (same N)
    if (lane == 0) red[0] = s;
  }
  __syncthreads();
  float total = red[0];
  __syncthreads();
  return total;
}

// Deterministic block max (wave shfl tree + 16-wave LDS combine).
__device__ __forceinline__ float blockMax(float v, float* red, int tid) {
#pragma unroll
  for (int m = 16; m >= 1; m >>= 1) v = fmaxf(v, __shfl_xor(v, m, 32));
  if ((tid & 31) == 0) red[tid >> 5] = v;
  __syncthreads();
  if (tid < 32) {
    float x = (tid < 16) ? red[tid] : -INFINITY;
#pragma unroll
    for (int m = 16; m >= 1; m >>= 1) x = fmaxf(x, __shfl_xor(x, m, 32));
    if (tid == 0) red[0] = x;
  }
  __syncthreads();
  float r = red[0];
  __syncthreads();
  return r;
}

// ---------------------------------------------------------------------------
// Kernel A: large groups (C > 500). grid = (1024 rows, 5 groups), 512 threads.
// ---------------------------------------------------------------------------
__global__ void __launch_bounds__(512) kd_large_kernel(
    const float* __restrict__ slog, const float* __restrict__ tlog,
    float* __restrict__ ws) {
  __shared__ float lt[8192];
  __shared__ float red[512];
  __shared__ int hist[256];
  __shared__ unsigned sh_prefix;
  __shared__ int sh_krem;

  const int gsz[5] = {8192, 4096, 2048, 1024, 512};
  const int goff[5] = {0, 8192, 12288, 14336, 15360};

  const int row = blockIdx.x;
  const int g = blockIdx.y;
  const int C = gsz[g];
  const int off = goff[g];
  const int tid = threadIdx.x;
  const int bd = 512;

  const float* trow = tlog + (size_t)row * TOT + off;
  const float* srow = slog + (size_t)row * TOT + off;

  // gfx1250 global_prefetch_b8: warm the student row while we stage teacher.
  __builtin_prefetch(srow + tid, 0, 0);

  for (int i = tid; i < C; i += bd) lt[i] = trow[i];
  __syncthreads();

  // ---- 4-pass radix select: theta = key of the 500th-largest teacher ----
  unsigned prefix = 0;
  int krem = KCAP;
  for (int shift = 24; shift >= 0; shift -= 8) {
    for (int i = tid; i < 256; i += bd) hist[i] = 0;
    __syncthreads();
    const unsigned himask = (shift == 24) ? 0u : (0xFFFFFFFFu << (shift + 8));
    for (int i = tid; i < C; i += bd) {
      unsigned k = fkey(lt[i]);
      if ((k & himask) == (prefix & himask))
        atomicAdd(&hist[(k >> shift) & 0xFF], 1);
    }
    __syncthreads();
    if (tid == 0) {
      int cum = 0, b = 255;
      for (;; --b) {
        int h = hist[b];
        if (cum + h >= krem || b == 0) break;
        cum += h;
      }
      sh_prefix = prefix | ((unsigned)b << shift);
      sh_krem = krem - cum;
    }
    __syncthreads();
    prefix = sh_prefix;
    krem = sh_krem;
    __syncthreads();
  }
  const unsigned theta = prefix;  // selected set = { key >= theta } (|.|=500)

  // ---- teacher max (top-500 contains the row max) ----
  float mloc = -INFINITY;
  for (int i = tid; i < C; i += bd) mloc = fmaxf(mloc, lt[i]);
  const float mt = blockMax(mloc, red, tid);

  // ---- fused pass: teacher from LDS, student read once from HBM ----
  // KL = (A1 - A2 + ms*Zt)/Zt - log Zt + log Zs, with
  //   A1 = sum e_t*(t-mt), A2 = sum e_t*s, Zt = sum e_t, Zs = sum e^(s-ms).
  float zt = 0.f, a1 = 0.f, a2 = 0.f;
  float msl = -INFINITY, zsl = 0.f;  // per-thread online softmax state
  for (int i = tid; i < C; i += bd) {
    float tv = lt[i];
    float sv = srow[i];  // unconditional -> coalesced
    if (fkey(tv) >= theta) {
      float td = tv - mt;
      float et = expf(td);
      zt += et;
      a1 += et * td;
      a2 += et * sv;
      if (sv > msl) {
        zsl = zsl * expf(msl - sv) + 1.0f;  // expf(-inf)=0 on first hit
        msl = sv;
      } else {
        zsl += expf(sv - msl);
      }
    }
  }
  const float ms = blockMax(msl, red, tid);
  float zadj = (msl == -INFINITY) ? 0.f : zsl * expf(msl - ms);
  const float Zs = blockSumWMMA(zadj, red, tid);
  const float Zt = blockSumWMMA(zt, red, tid);
  const float A1 = blockSumWMMA(a1, red, tid);
  const float A2 = blockSumWMMA(a2, red, tid);

  if (tid == 0) {
    float s1 = A1 - A2 + ms * Zt;
    float kl = s1 / Zt - logf(Zt) + logf(Zs);
    ws[row * 5 + g] = kl * NORM_OVER_B;
  }
}

// ---------------------------------------------------------------------------
// Kernel B: small groups (C <= 500 => k = C, full-group KL). One wave (32
// lanes) per group, 5 waves per row-block. Pure register + shfl reductions.
// ---------------------------------------------------------------------------
__global__ void __launch_bounds__(160) kd_small_kernel(
    const float* __restrict__ slog, const float* __restrict__ tlog,
    float* __restrict__ ws) {
  const int gsz[5] = {256, 128, 64, 32, 16};
  const int goff[5] = {15872, 16128, 16256, 16320, 16352};

  const int row = blockIdx.x;
  const int g = threadIdx.x >> 5;
  const int lane = threadIdx.x & 31;
  const int C = gsz[g];
  const int off = goff[g];

  const float* trow = tlog + (size_t)row * TOT + off;
  const float* srow = slog + (size_t)row * TOT + off;

  float tv[8], sv[8];
  int n = 0;
  float mt = -INFINITY, ms = -INFINITY;
  for (int i = lane; i < C; i += 32) {
    tv[n] = trow[i];
    sv[n] = srow[i];
    mt = fmaxf(mt, tv[n]);
    ms = fmaxf(ms, sv[n]);
    ++n;
  }
#pragma unroll
  for (int m = 16; m >= 1; m >>= 1) {
    mt = fmaxf(mt, __shfl_xor(mt, m, 32));
    ms = fmaxf(ms, __shfl_xor(ms, m, 32));
  }
  float zt = 0.f, zs = 0.f, s1 = 0.f;
  for (int c = 0; c < n; ++c) {
    float a = tv[c] - mt;
    float b = sv[c] - ms;
    float e = expf(a);
    zt += e;
    zs += expf(b);
    s1 += e * (a - b);
  }
#pragma unroll
  for (int m = 16; m >= 1; m >>= 1) {
    zt += __shfl_xor(zt, m, 32);
    zs += __shfl_xor(zs, m, 32);
    s1 += __shfl_xor(s1, m, 32);
  }
  if (lane == 0) {
    float kl = s1 / zt - logf(zt) + logf(zs);
    ws[5120 + row * 5 + g] = kl * NORM_OVER_B;
  }
}

// ---------------------------------------------------------------------------
// Kernel C: deterministic final sum of the 10240 partials (WMMA reducer).
// ---------------------------------------------------------------------------
__global__ void __launch_bounds__(512) kd_final_kernel(
    const float* __restrict__ ws, float* __restrict__ out) {
  __shared__ float red[512];
  const int tid = threadIdx.x;
  float acc = 0.f;
  for (int i = tid; i < NPART; i += 512) acc += ws[i];
  float total = blockSumWMMA(acc, red, tid);
  if (tid == 0) out[0] = total;
}

// ---------------------------------------------------------------------------
extern "C" void kernel_launch(void* const* d_in, const int* in_sizes, int n_in,
                              void* d_out, int out_size, void* d_ws,
                              size_t ws_size, hipStream_t stream) {
  (void)in_sizes;
  (void)n_in;
  (void)out_size;
  (void)ws_size;
  const float* slog = (const float*)d_in[0];  // student logits [1024,16368] f32
  const float* tlog = (const float*)d_in[1];  // teacher logits [1024,16368] f32
  // d_in[2] (targets, int64) is dead code in the reference.
  float* ws = (float*)d_ws;   // 10240 f32 partials; fully rewritten each call
  float* out = (float*)d_out; // 1 f32

  kd_large_kernel<<<dim3(1024, 5), 512, 0, stream>>>(slog, tlog, ws);
  kd_small_kernel<<<dim3(1024, 1), 160, 0, stream>>>(slog, tlog, ws);
  kd_final_kernel<<<1, 512, 0, stream>>>(ws, out);
}